// SLPLogitReg_82360292868174
// MI455X (gfx1250) — compile-verified
//
#include <hip/hip_runtime.h>
#include <hip/hip_bf16.h>
#include <math.h>

#define C_DIM 512
#define B_DIM 128
#define EPS_F 1e-6f
#define LAM_F (1.0f/3.0f)
#define BETA_F 0.7f

typedef float v2f __attribute__((ext_vector_type(2)));
typedef float v8f __attribute__((ext_vector_type(8)));

// coef_cc for a given class c
__device__ __forceinline__ float coef_cc_of(int c, const float* __restrict__ prop,
                                            const float* __restrict__ nz,
                                            const float* __restrict__ zv) {
    float ratio = nz[c] / (zv[c] + EPS_F);
    return (1.0f - (1.0f - LAM_F) * BETA_F) * prop[c] + (1.0f - LAM_F) * BETA_F * ratio;
}

__device__ __forceinline__ float coef_blend(float s, float r, float t) {
    return LAM_F * t + (1.0f - LAM_F) * (BETA_F * s + (1.0f - BETA_F) * r);
}

// Compute 4 contiguous coef values for row c, columns jbase..jbase+3 (branchless diagonal).
__device__ __forceinline__ void coef4_at(int c, int jbase, float ccc,
                                         const float4 sv, const float4 rv, const float4 tv,
                                         float out[4]) {
    const float s4[4] = {sv.x, sv.y, sv.z, sv.w};
    const float r4[4] = {rv.x, rv.y, rv.z, rv.w};
    const float t4[4] = {tv.x, tv.y, tv.z, tv.w};
    #pragma unroll
    for (int u = 0; u < 4; ++u) {
        const float v = coef_blend(s4[u], r4[u], t4[u]);
        out[u] = (jbase + u == c) ? ccc : v;   // v_cndmask, no EXEC games
    }
}

// ---------------- K1: row sums of coef via WMMA f32 16x16x4 (B = ones) ----------------
// grid = C/16 blocks, 32 threads (one wave32). Each wave reduces 16 rows.
// Branchless coef + float2 loads (j0 is even -> 8B aligned).
__global__ __launch_bounds__(32) void k1_rowsums(
        const float* __restrict__ sg, const float* __restrict__ ro, const float* __restrict__ ta,
        const float* __restrict__ prop, const float* __restrict__ nz, const float* __restrict__ zv,
        float* __restrict__ rowsum) {
    const int lane = threadIdx.x;
    const int half = lane >> 4;        // 0: K={0,1}, 1: K={2,3}
    const int m    = lane & 15;        // matrix row within tile
    const int row  = blockIdx.x * 16 + m;
    const float ccc = coef_cc_of(row, prop, nz, zv);

    const float2* __restrict__ sg2 = reinterpret_cast<const float2*>(sg);
    const float2* __restrict__ ro2 = reinterpret_cast<const float2*>(ro);
    const float2* __restrict__ ta2 = reinterpret_cast<const float2*>(ta);
    const int rowbase2 = (row * C_DIM) >> 1;

    v8f acc = {};
    v2f ones; ones.x = 1.0f; ones.y = 1.0f;
    for (int kt = 0; kt < C_DIM / 4; ++kt) {
        const int j0 = kt * 4 + half * 2;
        const int i2 = rowbase2 + (j0 >> 1);
        const float2 sv = sg2[i2];
        const float2 rv = ro2[i2];
        const float2 tv = ta2[i2];
        v2f a;
        const float v0 = coef_blend(sv.x, rv.x, tv.x);
        const float v1 = coef_blend(sv.y, rv.y, tv.y);
        a.x = (j0     == row) ? ccc : v0;
        a.y = (j0 + 1 == row) ? ccc : v1;
        acc = __builtin_amdgcn_wmma_f32_16x16x4_f32(
                  false, a, false, ones, (short)0, acc, false, false);
    }
    // D layout: lanes 0-15 vgpr r = D[M=r][N=lane]; lanes 16-31 vgpr r = D[M=8+r][N=lane-16].
    // B==ones => every N column holds the row sums. Lanes 0 and 16 write rows 0-7 / 8-15.
    if ((lane & 15) == 0) {
        const int base = blockIdx.x * 16 + half * 8;
        #pragma unroll
        for (int r = 0; r < 8; ++r) rowsum[base + r] = acc[r];
    }
}

// ---------------- K2: quant = sum(rowsum) / C^2 ----------------
__global__ __launch_bounds__(256) void k2_quant(const float* __restrict__ rowsum,
                                                float* __restrict__ stats) {
    __shared__ float s[256];
    float v = rowsum[threadIdx.x] + rowsum[threadIdx.x + 256];
    s[threadIdx.x] = v;
    __syncthreads();
    for (int o = 128; o > 0; o >>= 1) {
        if (threadIdx.x < (unsigned)o) s[threadIdx.x] += s[threadIdx.x + o];
        __syncthreads();
    }
    if (threadIdx.x == 0) stats[0] = s[0] / ((float)C_DIM * (float)C_DIM);
}

// ---------------- K3: per-row head/tail min/max/count partials ----------------
// grid = C blocks, 128 threads; each thread owns 4 contiguous columns (b128 loads).
// part layout: [hmin|hmax|hcnt|tmin|tmax|tcnt] x 512
__global__ __launch_bounds__(128) void k3_partials(
        const float* __restrict__ sg, const float* __restrict__ ro, const float* __restrict__ ta,
        const float* __restrict__ prop, const float* __restrict__ nz, const float* __restrict__ zv,
        const float* __restrict__ stats, float* __restrict__ part) {
    const int c = blockIdx.x, t = threadIdx.x;
    const float quant = stats[0];
    const float ccc = coef_cc_of(c, prop, nz, zv);

    const float4* __restrict__ sg4 = reinterpret_cast<const float4*>(sg + c * C_DIM);
    const float4* __restrict__ ro4 = reinterpret_cast<const float4*>(ro + c * C_DIM);
    const float4* __restrict__ ta4 = reinterpret_cast<const float4*>(ta + c * C_DIM);

    float vals[4];
    coef4_at(c, t * 4, ccc, sg4[t], ro4[t], ta4[t], vals);

    float hmin = 1e30f, hmax = -1e30f, hc = 0.0f;
    float tmin = 1e30f, tmax = -1e30f, tc = 0.0f;
    #pragma unroll
    for (int u = 0; u < 4; ++u) {
        const float v = vals[u];
        if (v > quant) { hmin = fminf(hmin, v); hmax = fmaxf(hmax, v); hc += 1.0f; }
        else           { tmin = fminf(tmin, v); tmax = fmaxf(tmax, v); tc += 1.0f; }
    }
    __shared__ float sh[6][128];
    sh[0][t] = hmin; sh[1][t] = hmax; sh[2][t] = hc;
    sh[3][t] = tmin; sh[4][t] = tmax; sh[5][t] = tc;
    __syncthreads();
    for (int o = 64; o > 0; o >>= 1) {
        if (t < o) {
            sh[0][t] = fminf(sh[0][t], sh[0][t + o]);
            sh[1][t] = fmaxf(sh[1][t], sh[1][t + o]);
            sh[2][t] += sh[2][t + o];
            sh[3][t] = fminf(sh[3][t], sh[3][t + o]);
            sh[4][t] = fmaxf(sh[4][t], sh[4][t + o]);
            sh[5][t] += sh[5][t + o];
        }
        __syncthreads();
    }
    if (t == 0) {
        part[0 * C_DIM + c] = sh[0][0];
        part[1 * C_DIM + c] = sh[1][0];
        part[2 * C_DIM + c] = sh[2][0];
        part[3 * C_DIM + c] = sh[3][0];
        part[4 * C_DIM + c] = sh[4][0];
        part[5 * C_DIM + c] = sh[5][0];
    }
}

// ---------------- K4: reduce partials -> global group stats ----------------
__global__ __launch_bounds__(256) void k4_stats(const float* __restrict__ part,
                                                float* __restrict__ stats) {
    const int t = threadIdx.x;
    __shared__ float sh[6][256];
    sh[0][t] = fminf(part[0 * C_DIM + t], part[0 * C_DIM + t + 256]);
    sh[1][t] = fmaxf(part[1 * C_DIM + t], part[1 * C_DIM + t + 256]);
    sh[2][t] = part[2 * C_DIM + t] + part[2 * C_DIM + t + 256];
    sh[3][t] = fminf(part[3 * C_DIM + t], part[3 * C_DIM + t + 256]);
    sh[4][t] = fmaxf(part[4 * C_DIM + t], part[4 * C_DIM + t + 256]);
    sh[5][t] = part[5 * C_DIM + t] + part[5 * C_DIM + t + 256];
    __syncthreads();
    for (int o = 128; o > 0; o >>= 1) {
        if (t < o) {
            sh[0][t] = fminf(sh[0][t], sh[0][t + o]);
            sh[1][t] = fmaxf(sh[1][t], sh[1][t + o]);
            sh[2][t] += sh[2][t + o];
            sh[3][t] = fminf(sh[3][t], sh[3][t + o]);
            sh[4][t] = fmaxf(sh[4][t], sh[4][t + o]);
            sh[5][t] += sh[5][t + o];
        }
        __syncthreads();
    }
    if (t == 0) {
        stats[1] = sh[0][0]; stats[2] = sh[1][0]; stats[3] = sh[2][0];
        stats[4] = sh[3][0]; stats[5] = sh[4][0]; stats[6] = sh[5][0];
    }
}

// ---------------- K5: per-row 23-bin step histograms -> weights ----------------
// grid = C blocks, 128 threads; 4 contiguous columns per thread (b128 loads).
// wts[c*24 + m]: m 0..9 = dw bins, 10..22 = up bins, 23 = always 0 (pad).
__global__ __launch_bounds__(128) void k5_weights(
        const float* __restrict__ sg, const float* __restrict__ ro, const float* __restrict__ ta,
        const float* __restrict__ prop, const float* __restrict__ nz, const float* __restrict__ zv,
        const float* __restrict__ stats, float* __restrict__ wts) {
    const int c = blockIdx.x, t = threadIdx.x;
    __shared__ int hist[24];
    if (t < 24) hist[t] = 0;
    __syncthreads();
    const float quant = stats[0];
    const float hmin = stats[1], hmax = stats[2], hc = stats[3];
    const float tmin = stats[4], tmax = stats[5], tc = stats[6];
    const float hden = hmax - hmin + EPS_F;
    const float tden = tmax - tmin + EPS_F;
    const float ccc = coef_cc_of(c, prop, nz, zv);

    const float4* __restrict__ sg4 = reinterpret_cast<const float4*>(sg + c * C_DIM);
    const float4* __restrict__ ro4 = reinterpret_cast<const float4*>(ro + c * C_DIM);
    const float4* __restrict__ ta4 = reinterpret_cast<const float4*>(ta + c * C_DIM);

    float vals[4];
    coef4_at(c, t * 4, ccc, sg4[t], ro4[t], ta4[t], vals);

    #pragma unroll
    for (int u = 0; u < 4; ++u) {
        const float v = vals[u];
        const bool split = v > quant;
        const float hx = split ? v : 0.0f;
        const float tx = split ? 0.0f : v;
        const float hcoef = (hc > 0.0f) ? fmaxf((hx - hmin) / hden, 0.0f) : hx;
        const float tcoef = (tc > 0.0f) ? fmaxf((tx - tmin) / tden, 0.0f) : tx;
        int sd = (int)floorf(hcoef * 9.0f);  sd = sd < 0 ? 0 : (sd > 9 ? 9 : sd);
        int su = (int)floorf(tcoef * 12.0f); su = su < 0 ? 0 : (su > 12 ? 12 : su);
        atomicAdd(&hist[sd], 1);
        atomicAdd(&hist[10 + su], 1);
    }
    __syncthreads();
    if (t < 24) wts[c * 24 + t] = (float)hist[t];
}

// ---------------- K6: per-(b,c) sigmoid chains + weighted sum ----------------
__global__ __launch_bounds__(256) void k6_output(
        const float* __restrict__ logits, const float* __restrict__ labels,
        const float* __restrict__ wts, float* __restrict__ out) {
    const int g = blockIdx.x * 256 + threadIdx.x;   // 0 .. B*C-1
    const int c = g & (C_DIM - 1);
    const float x0 = logits[g];
    const float y  = labels[g];

    // 24-float weight row; c*96 bytes is 16B aligned -> six b128 loads
    const float4* __restrict__ w4 = reinterpret_cast<const float4*>(wts + c * 24);
    float wl[24];
    #pragma unroll
    for (int q = 0; q < 6; ++q) {
        const float4 wv = w4[q];
        wl[q * 4 + 0] = wv.x; wl[q * 4 + 1] = wv.y;
        wl[q * 4 + 2] = wv.z; wl[q * 4 + 3] = wv.w;
    }

    const float inv_c = 1.0f / (float)C_DIM;
    float xd = x0, xu = x0, acc = 0.0f;
    #pragma unroll
    for (int m = 0; m < 10; ++m) {
        const float s = 1.0f / (1.0f + __expf(-xd));
        xd -= (s - y) * inv_c;
        acc += wl[m] * xd;
    }
    #pragma unroll
    for (int m = 0; m < 13; ++m) {
        const float s = 1.0f / (1.0f + __expf(-xu));
        xu += (s - y) * inv_c;
        acc += wl[10 + m] * xu;
    }
    out[g] = acc * inv_c - x0;
}

extern "C" void kernel_launch(void* const* d_in, const int* in_sizes, int n_in,
                              void* d_out, int out_size, void* d_ws, size_t ws_size,
                              hipStream_t stream) {
    const float* logits = (const float*)d_in[0];   // [B,C]
    const float* labels = (const float*)d_in[1];   // [B,C]
    const float* prop   = (const float*)d_in[2];   // [C]
    const float* nz     = (const float*)d_in[3];   // [C]
    const float* zv     = (const float*)d_in[4];   // [C]
    const float* sg     = (const float*)d_in[5];   // [C,C] normalized_sigma_cj
    const float* ro     = (const float*)d_in[6];   // [C,C] normalized_ro_cj
    const float* ta     = (const float*)d_in[7];   // [C,C] normalized_tao_cj
    float* out = (float*)d_out;

    // workspace layout (floats): rowsum[512] | stats[8]@512 | part[6*512]@1024 | wts[512*24]@4096
    float* ws      = (float*)d_ws;
    float* rowsum  = ws;
    float* stats   = ws + 512;
    float* part    = ws + 1024;
    float* wts     = ws + 4096;

    k1_rowsums<<<C_DIM / 16, 32, 0, stream>>>(sg, ro, ta, prop, nz, zv, rowsum);
    k2_quant<<<1, 256, 0, stream>>>(rowsum, stats);
    k3_partials<<<C_DIM, 128, 0, stream>>>(sg, ro, ta, prop, nz, zv, stats, part);
    k4_stats<<<1, 256, 0, stream>>>(part, stats);
    k5_weights<<<C_DIM, 128, 0, stream>>>(sg, ro, ta, prop, nz, zv, stats, wts);
    k6_output<<<(B_DIM * C_DIM) / 256, 256, 0, stream>>>(logits, labels, wts, out);
}